// PrimaryCaps_4054449127615
// MI455X (gfx1250) — compile-verified
//
#include <hip/hip_runtime.h>
#include <hip/hip_bf16.h>

typedef __attribute__((ext_vector_type(2))) float v2f;
typedef __attribute__((ext_vector_type(8))) float v8f;

#define C_IN        32
#define O_CH        544             // 512 pose + 32 act = 34 tiles of 16
#define NB          64
#define HW          3136            // 56*56 = 196 column tiles of 16
#define CT_PER_N    (HW / 16)       // 196 (divisible by TILES_PW -> n uniform per wave)
#define TOTAL_CT    (NB * CT_PER_N) // 12544
#define WAVES_PB    8
#define TILES_PW    4               // column tiles per wave

// out[n, o, hw] = sum_c Wc[o, c] * x[n, c, hw] + bc[o]
// One wave computes a 16(o) x 16(hw) tile via 8x V_WMMA_F32_16X16X4_F32 (K=32),
// split into two independent accumulation chains for matrix-pipe ILP.
__global__ __launch_bounds__(256) void primarycaps_f32_wmma_kernel(
    const float* __restrict__ x,   // [N, 32, HW]
    const float* __restrict__ Wp,  // [512, 32]
    const float* __restrict__ bp,  // [512]
    const float* __restrict__ Wa,  // [32, 32]
    const float* __restrict__ ba,  // [32]
    float* __restrict__ out)       // [N, 544, HW]
{
    const int lane  = threadIdx.x & 31;
    const int wave  = threadIdx.x >> 5;
    const int otile = blockIdx.x;            // 0..33 (0..31 pose, 32..33 act)
    const int m_lo  = lane & 15;             // A row (M) / B,D column (N)
    const int khalf = lane >> 4;             // 0: K+{0,1}, 1: K+{2,3}

    // ---- A tile: Wc[otile*16 + m, 0..31], hoisted out of the spatial loop ----
    const int o_row = otile * 16 + m_lo;
    const float* wrow = (o_row < 512) ? (Wp + o_row * C_IN)
                                      : (Wa + (o_row - 512) * C_IN);
    v2f a[8];
#pragma unroll
    for (int c = 0; c < 8; ++c) {
        const int k = c * 4 + khalf * 2;
        a[c].x = wrow[k];
        a[c].y = wrow[k + 1];
    }

    // ---- bias for the 8 D rows this lane owns: M = r + khalf*8 ----
    float bias[8];
#pragma unroll
    for (int r = 0; r < 8; ++r) {
        const int o = otile * 16 + khalf * 8 + r;
        bias[r] = (o < 512) ? bp[o] : ba[o - 512];
    }

    // ---- base pointers: n is constant across this wave's 4 tiles ----
    const int ct0 = (blockIdx.y * WAVES_PB + wave) * TILES_PW;
    const int n   = ct0 / CT_PER_N;
    const int ct  = ct0 - n * CT_PER_N;
    const int col = ct * 16 + m_lo;

    const float* xb = x + n * C_IN * HW + col;                       // + k*HW per k
    float*       ob = out + (n * O_CH + otile * 16 + khalf * 8) * HW + col;

#pragma unroll 1
    for (int t = 0; t < TILES_PW; ++t) {
        // load the full 32x16 B tile (16 dwords/lane, clause-coalesced)
        v2f b[8];
#pragma unroll
        for (int c = 0; c < 8; ++c) {
            const int k = c * 4 + khalf * 2;
            b[c].x = xb[k * HW];
            b[c].y = xb[(k + 1) * HW];
        }

        // two independent K-chains: even chunks -> acc0, odd chunks -> acc1
        v8f acc0 = {}, acc1 = {};
#pragma unroll
        for (int c = 0; c < 8; c += 2) {
            acc0 = __builtin_amdgcn_wmma_f32_16x16x4_f32(
                false, a[c],     false, b[c],     (short)0, acc0, false, false);
            acc1 = __builtin_amdgcn_wmma_f32_16x16x4_f32(
                false, a[c + 1], false, b[c + 1], (short)0, acc1, false, false);
        }

        // combine chains + bias; rows M = khalf*8 + r, non-temporal (write-once)
#pragma unroll
        for (int r = 0; r < 8; ++r) {
            __builtin_nontemporal_store(acc0[r] + acc1[r] + bias[r], ob + r * HW);
        }

        xb += 16;   // next 16 spatial columns
        ob += 16;
    }
}

extern "C" void kernel_launch(void* const* d_in, const int* in_sizes, int n_in,
                              void* d_out, int out_size, void* d_ws, size_t ws_size,
                              hipStream_t stream) {
    const float* x  = (const float*)d_in[0];
    const float* Wp = (const float*)d_in[1];
    const float* bp = (const float*)d_in[2];
    const float* Wa = (const float*)d_in[3];
    const float* ba = (const float*)d_in[4];
    float* out = (float*)d_out;

    dim3 grid(O_CH / 16, TOTAL_CT / (WAVES_PB * TILES_PW));  // (34, 392)
    primarycaps_f32_wmma_kernel<<<grid, 256, 0, stream>>>(x, Wp, bp, Wa, ba, out);
}